// PdhgTgvTorch_87282325389598
// MI455X (gfx1250) — compile-verified
//
#include <hip/hip_runtime.h>
#include <stdint.h>

// ---------------------------------------------------------------------------
// PDHG-TGV solver, 1024x1024, 32 iterations, fp32.
// Whole state (44 MB) is L2-resident on MI455X (192 MB L2) -> L2-BW bound.
// Both stencil kernels stage their tiles with the CDNA5 Tensor Data Mover
// (one tensor_load_to_lds per plane, completion on TENSORcnt); per-lane
// async global->LDS loads are the compile-time fallback. w is symmetric ->
// stored as 3 planes (w00, w01, w11). All updates are in-place (pointwise in
// the written field), so no double buffering is needed.
// ---------------------------------------------------------------------------

#define NDIM   1024
#define TI     16            // tile rows  (i / axis 0)
#define TJ     64            // tile cols  (j / axis 1)
#define RH     (TI + 1)      // staged region height (with 1 halo)
#define RW     (TJ + 1)      // staged region width  (with 1 halo)
#define REG    (RH * RW)     // 1105 elements per staged plane
#define NT     256           // threads per block (8 waves, wave32)
#define SIGMA  0.29f
#define TAU    0.29f
#define NITERS 32

#if defined(__has_builtin)
#if __has_builtin(__builtin_amdgcn_tensor_load_to_lds) && \
    __has_builtin(__builtin_amdgcn_s_wait_tensorcnt)
#define HAVE_TDM 1
#endif
#endif
#ifndef HAVE_TDM
#define HAVE_TDM 0
#endif

// ---- CDNA5 async global->LDS copy (fallback path) -------------------------
__device__ __forceinline__ void async_ld_f32(uint32_t lds_addr, const float* g) {
  asm volatile("global_load_async_to_lds_b32 %0, %1, off"
               :: "v"(lds_addr), "v"(g) : "memory");
}
__device__ __forceinline__ void wait_async0() {
  asm volatile("s_wait_asynccnt 0" ::: "memory");
}

__device__ __forceinline__ int clampN(int x) {
  return x < 0 ? 0 : (x > NDIM - 1 ? NDIM - 1 : x);
}

// Stage an RH x RW region of plane g (rows ib.., cols jb.., clamped) into LDS
// using async loads. Boundary semantics are applied at compute time via
// select masks, so clamped loads are safe placeholders.
__device__ __forceinline__ void stage(float* s, const float* __restrict__ g,
                                      int ib, int jb, int tid) {
  for (int e = tid; e < REG; e += NT) {
    const int ri = e / RW;
    const int rj = e - ri * RW;
    const int gi = clampN(ib + ri);
    const int gj = clampN(jb + rj);
    // low 32 bits of a generic pointer to __shared__ == wave-relative LDS addr
    async_ld_f32((uint32_t)(uintptr_t)(s + e), g + (size_t)gi * NDIM + gj);
  }
}

#if HAVE_TDM
// ---- CDNA5 Tensor Data Mover: one-instruction 65x17 f32 tile DMA ----------
typedef uint32_t tdm_v4u __attribute__((ext_vector_type(4)));
typedef int      tdm_v8i __attribute__((ext_vector_type(8)));
typedef int      tdm_v4i __attribute__((ext_vector_type(4)));

// D# per CDNA5 ISA:
//  group0: [1:0]=count=1, [63:32]=lds_addr, [120:64]=global_addr, [127:126]=2
//  group1: [17:16]=data_size(2 -> 4B), [79:48]=tensor_dim0=1024,
//          [111:80]=tensor_dim1=1024, [127:112]=tile_dim0=65,
//          [143:128]=tile_dim1=17, [207:160]=tensor_dim0_stride=1024
// LDS fill is contiguous (no pad) -> matches s[ri*RW + rj] layout exactly.
__device__ __forceinline__ void tdm_load_tile(float* s, const float* gtile) {
  const uint64_t ga = (uint64_t)(uintptr_t)gtile;
  tdm_v4u g0;
  g0.x = 1u;                                               // count=1, user mode
  g0.y = (uint32_t)(uintptr_t)s;                           // LDS byte address
  g0.z = (uint32_t)(ga & 0xFFFFFFFFu);                     // global_addr[31:0]
  g0.w = (uint32_t)((ga >> 32) & 0x1FFFFFFu) | (2u << 30); // addr[56:32]|type=2
  tdm_v8i g1;
  g1[0] = 0x20000;                 // data_size = 4 bytes
  g1[1] = (int)(NDIM << 16);       // tensor_dim0[15:0] in [63:48]
  g1[2] = (int)(NDIM << 16);       // tensor_dim0 hi=0 | tensor_dim1[15:0]
  g1[3] = (int)(RW << 16);         // tensor_dim1 hi=0 | tile_dim0=65
  g1[4] = RH;                      // tile_dim1=17, tile_dim2=0
  g1[5] = NDIM;                    // tensor_dim0_stride = 1024 elements
  g1[6] = 0;
  g1[7] = 0;
  const tdm_v4i gz4 = {0, 0, 0, 0};               // groups 2/3 unused (2-D)
#if __clang_major__ >= 23
  const tdm_v8i gz8 = {0, 0, 0, 0, 0, 0, 0, 0};   // extra group (clang-23 ABI)
  __builtin_amdgcn_tensor_load_to_lds(g0, g1, gz4, gz4, gz8, 0);
#else
  __builtin_amdgcn_tensor_load_to_lds(g0, g1, gz4, gz4, 0);
#endif
}
#endif

// ---------------------------------------------------------------------------
// Init: deinterleave p, copy u, zero v/w, set u_bar=u, p_bar=p.
// ---------------------------------------------------------------------------
__global__ __launch_bounds__(NT)
void tgv_init(const float* __restrict__ u_in, const float* __restrict__ p_in,
              float* __restrict__ U, float* __restrict__ P0, float* __restrict__ P1,
              float* __restrict__ UB, float* __restrict__ PB0, float* __restrict__ PB1,
              float* __restrict__ V0, float* __restrict__ V1,
              float* __restrict__ W00, float* __restrict__ W01, float* __restrict__ W11) {
  const int idx = blockIdx.x * NT + threadIdx.x;
  const float u = u_in[idx];
  const float p0 = p_in[2 * idx];
  const float p1 = p_in[2 * idx + 1];
  U[idx] = u;  UB[idx] = u;
  P0[idx] = p0; P1[idx] = p1;
  PB0[idx] = p0; PB1[idx] = p1;
  V0[idx] = 0.f; V1[idx] = 0.f;
  W00[idx] = 0.f; W01[idx] = 0.f; W11[idx] = 0.f;
}

// ---------------------------------------------------------------------------
// Dual update:
//   v <- P_alpha1(v + sigma*(nabla(u_bar) - p_bar))        (forward diffs)
//   w <- P_alpha0(w + sigma*e(p_bar))                      (backward diffs)
// Pointwise in v and w -> in-place safe. Tiles staged by TDM (wave 0 issues
// the DMAs; TDM ignores EXEC). Halo cells loaded from outside the logical
// plane may hold neighbor-plane garbage but are never consumed: every
// boundary term is select-masked.
// ---------------------------------------------------------------------------
__global__ __launch_bounds__(NT)
void tgv_dual(const float* __restrict__ UB,
              const float* __restrict__ PB0, const float* __restrict__ PB1,
              float* __restrict__ V0, float* __restrict__ V1,
              float* __restrict__ W00, float* __restrict__ W01, float* __restrict__ W11,
              const float* __restrict__ A1, const float* __restrict__ A0) {
  __shared__ float su[REG];   // u_bar,  region origin (i0,   j0)   : halo down/right
  __shared__ float sp0[REG];  // p_bar0, region origin (i0-1, j0-1) : halo up/left
  __shared__ float sp1[REG];

  const int tid = threadIdx.x;
  const int j0 = blockIdx.x * TJ;
  const int i0 = blockIdx.y * TI;

#if HAVE_TDM
  if (tid < 32) {  // one wave issues the DMAs
    tdm_load_tile(su,  UB  + (ptrdiff_t)i0 * NDIM + j0);
    tdm_load_tile(sp0, PB0 + (ptrdiff_t)(i0 - 1) * NDIM + (j0 - 1));
    tdm_load_tile(sp1, PB1 + (ptrdiff_t)(i0 - 1) * NDIM + (j0 - 1));
    __builtin_amdgcn_s_wait_tensorcnt(0);
  }
#else
  stage(su,  UB,  i0,     j0,     tid);
  stage(sp0, PB0, i0 - 1, j0 - 1, tid);
  stage(sp1, PB1, i0 - 1, j0 - 1, tid);
  wait_async0();
#endif
  __syncthreads();

  const int jl = tid & (TJ - 1);
  const int ib = tid >> 6;          // 0..3
  const int gj = j0 + jl;

#pragma unroll
  for (int k = 0; k < 4; ++k) {
    const int il = ib * 4 + k;
    const int gi = i0 + il;
    const int idx = gi * NDIM + gj;

    // forward diffs of u_bar (zero at last index)
    const float uc  = su[il * RW + jl];
    const float dxu = (gi < NDIM - 1) ? (su[(il + 1) * RW + jl] - uc) : 0.f;
    const float dyu = (gj < NDIM - 1) ? (su[il * RW + (jl + 1)] - uc) : 0.f;

    // p_bar center/up/left from the shifted region
    const int c = (il + 1) * RW + (jl + 1);
    const float p0c = sp0[c], p1c = sp1[c];
    const float p0u = sp0[c - RW], p1u = sp1[c - RW];
    const float p0l = sp0[c - 1],  p1l = sp1[c - 1];

    // ---- v update ----
    float q0 = V0[idx] + SIGMA * (dxu - p0c);
    float q1 = V1[idx] + SIGMA * (dyu - p1c);
    const float nv = sqrtf(q0 * q0 + q1 * q1);
    const float fv = fmaxf(1.f, nv / A1[idx]);
    V0[idx] = q0 / fv;
    V1[idx] = q1 / fv;

    // ---- w update: backward diffs of p_bar ----
    // bx(t)[i] = (i==N-1 ? 0 : t[i]) - (i==0 ? 0 : t[i-1])  (matches jnp concat)
    const float d1x = ((gi == NDIM - 1) ? 0.f : p0c) - ((gi == 0) ? 0.f : p0u);
    const float d1y = ((gj == NDIM - 1) ? 0.f : p0c) - ((gj == 0) ? 0.f : p0l);
    const float d2x = ((gi == NDIM - 1) ? 0.f : p1c) - ((gi == 0) ? 0.f : p1u);
    const float d2y = ((gj == NDIM - 1) ? 0.f : p1c) - ((gj == 0) ? 0.f : p1l);
    const float off = 0.5f * (d1y + d2x);

    float a = W00[idx] + SIGMA * d1x;
    float b = W01[idx] + SIGMA * off;
    float cc = W11[idx] + SIGMA * d2y;
    const float nw = sqrtf(a * a + 2.f * b * b + cc * cc);
    const float fw = fmaxf(1.f, nw / A0[idx]);
    W00[idx] = a / fw;
    W01[idx] = b / fw;
    W11[idx] = cc / fw;
  }
}

// ---------------------------------------------------------------------------
// Primal update:
//   u <- (u + tau*(div_v(v) + f)) / (1+tau)      (backward diffs of v)
//   p <- p + tau*(v + div_w(w))                  (forward diffs of w)
//   u_bar <- 2u_new - u_old ; p_bar <- 2p_new - p_old
// Pointwise in u and p -> in-place safe. Tiles staged by TDM (same masking /
// guard-row argument as the dual kernel; W11 overreads land in the tail
// guard rows of the workspace).
// ---------------------------------------------------------------------------
__global__ __launch_bounds__(NT)
void tgv_primal(float* __restrict__ U, float* __restrict__ P0, float* __restrict__ P1,
                float* __restrict__ UB, float* __restrict__ PB0, float* __restrict__ PB1,
                const float* __restrict__ V0, const float* __restrict__ V1,
                const float* __restrict__ W00, const float* __restrict__ W01,
                const float* __restrict__ W11, const float* __restrict__ F) {
  __shared__ float sv0[REG];  // v,   region origin (i0-1, j0-1): halo up/left
  __shared__ float sv1[REG];
  __shared__ float sa[REG];   // w00, region origin (i0, j0): halo down/right
  __shared__ float sb[REG];   // w01
  __shared__ float sc[REG];   // w11

  const int tid = threadIdx.x;
  const int j0 = blockIdx.x * TJ;
  const int i0 = blockIdx.y * TI;

#if HAVE_TDM
  if (tid < 32) {  // one wave issues the DMAs
    tdm_load_tile(sv0, V0  + (ptrdiff_t)(i0 - 1) * NDIM + (j0 - 1));
    tdm_load_tile(sv1, V1  + (ptrdiff_t)(i0 - 1) * NDIM + (j0 - 1));
    tdm_load_tile(sa,  W00 + (ptrdiff_t)i0 * NDIM + j0);
    tdm_load_tile(sb,  W01 + (ptrdiff_t)i0 * NDIM + j0);
    tdm_load_tile(sc,  W11 + (ptrdiff_t)i0 * NDIM + j0);
    __builtin_amdgcn_s_wait_tensorcnt(0);
  }
#else
  stage(sv0, V0,  i0 - 1, j0 - 1, tid);
  stage(sv1, V1,  i0 - 1, j0 - 1, tid);
  stage(sa,  W00, i0,     j0,     tid);
  stage(sb,  W01, i0,     j0,     tid);
  stage(sc,  W11, i0,     j0,     tid);
  wait_async0();
#endif
  __syncthreads();

  const int jl = tid & (TJ - 1);
  const int ib = tid >> 6;
  const int gj = j0 + jl;
  const float inv1pt = 1.f / (1.f + TAU);

#pragma unroll
  for (int k = 0; k < 4; ++k) {
    const int il = ib * 4 + k;
    const int gi = i0 + il;
    const int idx = gi * NDIM + gj;

    // ---- u update: div_h_v = bx(v0) + by(v1) ----
    const int c = (il + 1) * RW + (jl + 1);
    const float v0c = sv0[c], v1c = sv1[c];
    const float v0u = sv0[c - RW];
    const float v1l = sv1[c - 1];
    const float divv = (((gi == NDIM - 1) ? 0.f : v0c) - ((gi == 0) ? 0.f : v0u))
                     + (((gj == NDIM - 1) ? 0.f : v1c) - ((gj == 0) ? 0.f : v1l));
    const float uo = U[idx];
    const float un = (uo + TAU * (divv + F[idx])) * inv1pt;

    // ---- p update: div_h_w, forward diffs of (a,b,c) ----
    const int cw = il * RW + jl;
    const float ac = sa[cw], bc = sb[cw], ccc = sc[cw];
    const float dxa = (gi < NDIM - 1) ? (sa[cw + RW] - ac)  : 0.f;
    const float dxb = (gi < NDIM - 1) ? (sb[cw + RW] - bc)  : 0.f;
    const float dyb = (gj < NDIM - 1) ? (sb[cw + 1] - bc)   : 0.f;
    const float dyc = (gj < NDIM - 1) ? (sc[cw + 1] - ccc)  : 0.f;
    const float p0o = P0[idx];
    const float p1o = P1[idx];
    const float p0n = p0o + TAU * (v0c + dxa + dyb);
    const float p1n = p1o + TAU * (v1c + dxb + dyc);

    U[idx] = un;   P0[idx] = p0n;   P1[idx] = p1n;
    UB[idx]  = 2.f * un  - uo;
    PB0[idx] = 2.f * p0n - p0o;
    PB1[idx] = 2.f * p1n - p1o;
  }
}

__global__ __launch_bounds__(NT)
void tgv_copy_out(const float* __restrict__ U, float* __restrict__ out) {
  const int idx = blockIdx.x * NT + threadIdx.x;
  out[idx] = U[idx];
}

// ---------------------------------------------------------------------------
extern "C" void kernel_launch(void* const* d_in, const int* in_sizes, int n_in,
                              void* d_out, int out_size, void* d_ws, size_t ws_size,
                              hipStream_t stream) {
  const float* u_in   = (const float*)d_in[0];   // also the data term f
  const float* p_in   = (const float*)d_in[1];   // interleaved [n,n,2]
  const float* alpha1 = (const float*)d_in[2];
  const float* alpha0 = (const float*)d_in[3];
  // d_in[4] = num_iters (device scalar, == 32 per reference); hardcoded NITERS.

  // Plane base is offset by one guard row so that TDM tile origins at
  // (i0-1, j0-1) stay inside the allocation; high-side overreads (row/col
  // 1024 spill) land in the next plane or in the 2*NDIM tail-guard floats
  // after the last plane. Overread values are never consumed (masked).
  // Workspace requirement: (11*N*N + 3*N) floats ~= 44.01 MB.
  float* ws = (float*)d_ws + NDIM;
  const size_t PL = (size_t)NDIM * NDIM;
  float* U   = ws + 0 * PL;
  float* P0  = ws + 1 * PL;
  float* P1  = ws + 2 * PL;
  float* UB  = ws + 3 * PL;
  float* PB0 = ws + 4 * PL;
  float* PB1 = ws + 5 * PL;
  float* V0  = ws + 6 * PL;
  float* V1  = ws + 7 * PL;
  float* W00 = ws + 8 * PL;
  float* W01 = ws + 9 * PL;
  float* W11 = ws + 10 * PL;

  const int nblk1d = (int)(PL / NT);
  tgv_init<<<nblk1d, NT, 0, stream>>>(u_in, p_in, U, P0, P1, UB, PB0, PB1,
                                      V0, V1, W00, W01, W11);

  dim3 grid(NDIM / TJ, NDIM / TI);  // 16 x 64 tiles
  for (int it = 0; it < NITERS; ++it) {
    tgv_dual<<<grid, NT, 0, stream>>>(UB, PB0, PB1, V0, V1, W00, W01, W11,
                                      alpha1, alpha0);
    tgv_primal<<<grid, NT, 0, stream>>>(U, P0, P1, UB, PB0, PB1,
                                        V0, V1, W00, W01, W11, u_in);
  }

  tgv_copy_out<<<nblk1d, NT, 0, stream>>>(U, (float*)d_out);
}